// Network_72808285602501
// MI455X (gfx1250) — compile-verified
//
#include <hip/hip_runtime.h>
#include <cmath>

typedef __attribute__((ext_vector_type(2))) float v2f;
typedef __attribute__((ext_vector_type(4))) float v4f;
typedef __attribute__((ext_vector_type(8))) float v8f;

#define UNITS 10
#define OUTN  4
#define BATCH 256
#define TLEN  2048
#define DIM   64
#define K2    (2 * UNITS)   // 20

// ---------------------------------------------------------------------------
// Phase 1: pre[m][n] = sum_k tx[m][k] * kernel[k][n] + bias[n]
//   m = b*TLEN + t  in [0, BATCH*TLEN),  k in [0,64),  n in [0,20) (pad to 32)
// One wave computes a 16(M) x 32(N) tile via 2 accumulators, 16 chained
// V_WMMA_F32_16X16X4_F32 per accumulator (K=64 = 16 chunks of 4).
// A-fragment layout (ISA 7.12.2, 32-bit A 16x4): lane l holds row M=l%16,
//   VGPR0 = K0+(l<16?0:2), VGPR1 = K0+(l<16?1:3)  -> one v2f load per chunk.
// B-fragment symmetric: lane l holds col N=n0+l%16 with same K striping.
// C/D layout: lane l, VGPR r -> M = r + 8*(l/16), N = l%16.
// ---------------------------------------------------------------------------
__global__ __launch_bounds__(256) void gemm_pre_wmma(
    const float* __restrict__ tx,    // [BATCH*TLEN][DIM]
    const float* __restrict__ kern,  // [DIM][K2]
    const float* __restrict__ bias,  // [K2]
    float* __restrict__ pre)         // [BATCH*TLEN][K2]
{
  const int lane   = threadIdx.x & 31;
  const int wave   = blockIdx.x * (blockDim.x >> 5) + (threadIdx.x >> 5);
  const int nwaves = gridDim.x * (blockDim.x >> 5);
  const int l16    = lane & 15;
  const int koff   = (lane >> 4) * 2;   // 0 for lanes 0-15, 2 for lanes 16-31

  // Preload B fragments for all 16 K-chunks, both N-tiles (invariant data).
  v2f b0[16], b1[16];
#pragma unroll
  for (int c = 0; c < 16; ++c) {
    const int k  = c * 4 + koff;
    const int n0 = l16;          // N-tile 0: cols 0..15 (all < 20)
    const int n1 = 16 + l16;     // N-tile 1: cols 16..31 (valid only < 20)
    b0[c].x = kern[k * K2 + n0];
    b0[c].y = kern[(k + 1) * K2 + n0];
    b1[c].x = (n1 < K2) ? kern[k * K2 + n1] : 0.0f;
    b1[c].y = (n1 < K2) ? kern[(k + 1) * K2 + n1] : 0.0f;
  }
  const float bias0 = bias[l16];
  const float bias1 = (16 + l16 < K2) ? bias[16 + l16] : 0.0f;

  const int MT = (BATCH * TLEN) / 16;   // 32768 M-tiles
  for (int mt = wave; mt < MT; mt += nwaves) {
    const int m0 = mt * 16;
    const float* arow = tx + (size_t)(m0 + l16) * DIM + koff;

    v8f c0 = {};
    v8f c1 = {};
#pragma unroll
    for (int c = 0; c < 16; ++c) {
      v2f a = *(const v2f*)(arow + c * 4);
      c0 = __builtin_amdgcn_wmma_f32_16x16x4_f32(false, a, false, b0[c],
                                                 (short)0, c0, false, false);
      c1 = __builtin_amdgcn_wmma_f32_16x16x4_f32(false, a, false, b1[c],
                                                 (short)0, c1, false, false);
    }

    // Store C: lane l, reg r -> row m0 + 8*(lane/16) + r, col l16 / 16+l16.
    const int mbase = m0 + (lane >> 4) * 8;
    const bool hi_ok = (16 + l16) < K2;
#pragma unroll
    for (int r = 0; r < 8; ++r) {
      float* prow = pre + (size_t)(mbase + r) * K2;
      prow[l16] = c0[r] + bias0;
      if (hi_ok) prow[16 + l16] = c1[r] + bias1;
    }
  }
}

// ---------------------------------------------------------------------------
// Phase 2: serial GRU-like scan, one lane per batch element (256 lanes =
// 8 wave32s). Recurrent weights live entirely in VGPRs (10x20 floats/lane);
// inner mat-vec is a fully unrolled 200-FMA chain with 20 independent
// accumulators. Folds the fc + softmax head at the end.
// ---------------------------------------------------------------------------
__global__ __launch_bounds__(32) void scan_head(
    const float* __restrict__ pre,    // [BATCH*TLEN][K2], m = b*TLEN + t
    const float* __restrict__ rec,    // [UNITS][K2]
    const float* __restrict__ fc_w,   // [UNITS][OUTN]
    const float* __restrict__ fc_b,   // [OUTN]
    float* __restrict__ out)          // [BATCH][OUTN]
{
  const int b = blockIdx.x * 32 + threadIdx.x;
  if (b >= BATCH) return;

  // Register-resident recurrent weights.
  float rk[UNITS][K2];
#pragma unroll
  for (int j = 0; j < UNITS; ++j)
#pragma unroll
    for (int k = 0; k < K2; ++k)
      rk[j][k] = rec[j * K2 + k];

  float vh[UNITS];
  float vs[UNITS];
#pragma unroll
  for (int i = 0; i < UNITS; ++i) { vh[i] = 0.0f; vs[i] = 0.0f; }

  const float* pb = pre + (size_t)b * TLEN * K2;

#pragma unroll 1
  for (int t = 0; t < TLEN; ++t) {
    const float* p = pb + (size_t)t * K2;
    // 80-byte row, 16B-aligned (80 % 16 == 0): 5 x b128 loads.
    const v4f q0 = *(const v4f*)(p + 0);
    const v4f q1 = *(const v4f*)(p + 4);
    const v4f q2 = *(const v4f*)(p + 8);
    const v4f q3 = *(const v4f*)(p + 12);
    const v4f q4 = *(const v4f*)(p + 16);
    // Pull next timestep toward the caches while we crunch this one.
    __builtin_prefetch(p + K2, 0, 1);

    float acc[K2];
    acc[0]=q0.x;  acc[1]=q0.y;  acc[2]=q0.z;  acc[3]=q0.w;
    acc[4]=q1.x;  acc[5]=q1.y;  acc[6]=q1.z;  acc[7]=q1.w;
    acc[8]=q2.x;  acc[9]=q2.y;  acc[10]=q2.z; acc[11]=q2.w;
    acc[12]=q3.x; acc[13]=q3.y; acc[14]=q3.z; acc[15]=q3.w;
    acc[16]=q4.x; acc[17]=q4.y; acc[18]=q4.z; acc[19]=q4.w;

#pragma unroll
    for (int j = 0; j < UNITS; ++j) {
      const float h = vh[j];
#pragma unroll
      for (int k = 0; k < K2; ++k)
        acc[k] = fmaf(h, rk[j][k], acc[k]);
    }

#pragma unroll
    for (int i = 0; i < UNITS; ++i) {
      const float v1 = 1.0f / (1.0f + __expf(-acc[i]));
      const float v2 = tanhf(acc[UNITS + i]);
      const float s  = v1 * vs[i] + (1.0f - v1) * v2;
      vs[i] = s;
      vh[i] = tanhf(s);
    }
  }

  // Head: logits = vh @ fc_w + fc_b ; softmax over 4.
  float logit[OUTN];
#pragma unroll
  for (int o = 0; o < OUTN; ++o) {
    float a = fc_b[o];
#pragma unroll
    for (int j = 0; j < UNITS; ++j)
      a = fmaf(vh[j], fc_w[j * OUTN + o], a);
    logit[o] = a;
  }
  float mx = fmaxf(fmaxf(logit[0], logit[1]), fmaxf(logit[2], logit[3]));
  float e[OUTN];
  float sum = 0.0f;
#pragma unroll
  for (int o = 0; o < OUTN; ++o) { e[o] = __expf(logit[o] - mx); sum += e[o]; }
  const float inv = 1.0f / sum;
#pragma unroll
  for (int o = 0; o < OUTN; ++o)
    out[(size_t)b * OUTN + o] = e[o] * inv;
}

// ---------------------------------------------------------------------------
extern "C" void kernel_launch(void* const* d_in, const int* in_sizes, int n_in,
                              void* d_out, int out_size, void* d_ws, size_t ws_size,
                              hipStream_t stream) {
  const float* tx   = (const float*)d_in[0];  // [256][2048][64]
  const float* kern = (const float*)d_in[1];  // [64][20]
  const float* rec  = (const float*)d_in[2];  // [10][20]
  const float* bias = (const float*)d_in[3];  // [20]
  const float* fc_w = (const float*)d_in[4];  // [10][4]
  const float* fc_b = (const float*)d_in[5];  // [4]
  float* out = (float*)d_out;                 // [256][4]
  float* pre = (float*)d_ws;                  // [256*2048][20] = 40 MiB scratch

  // Phase 1: WMMA GEMM. 1024 blocks x 8 waves = 8192 waves, 4 tiles each.
  gemm_pre_wmma<<<1024, 256, 0, stream>>>(tx, kern, bias, pre);
  // Phase 2: 256 lanes = 8 wave32 blocks spread across WGPs.
  scan_head<<<8, 32, 0, stream>>>(pre, rec, fc_w, fc_b, out);
}